// SiReN_77378130805157
// MI455X (gfx1250) — compile-verified
//
#include <hip/hip_runtime.h>
#include <math.h>

#define M_USERS   100000
#define N_ITEMS   50000
#define NUM_NODES (M_USERS + N_ITEMS)   // 150000
#define DIM       64
#define N_EDGES   3200000
#define BATCH     1024
#define N_NEGS    40
#define REG_C     1e-4f

typedef __attribute__((ext_vector_type(2))) float v2f;
typedef __attribute__((ext_vector_type(8))) float v8f;

// ---------------------------------------------------------------- utilities

__global__ void fill_zero_kernel(float* __restrict__ p, long long n) {
    long long i = (long long)blockIdx.x * blockDim.x + threadIdx.x;
    long long stride = (long long)gridDim.x * blockDim.x;
    for (; i < n; i += stride) p[i] = 0.0f;
}

// ------------------------------------------------- LightGCN scatter kernels

__global__ void deg_kernel(const int* __restrict__ col, float* __restrict__ deg) {
    int e = blockIdx.x * blockDim.x + threadIdx.x;
    if (e < N_EDGES) atomicAdd(&deg[col[e]], 1.0f);
}

__global__ void dinv_kernel(const float* __restrict__ deg, float* __restrict__ dinv) {
    int i = blockIdx.x * blockDim.x + threadIdx.x;
    if (i < NUM_NODES) {
        float d = deg[i];
        dinv[i] = (d > 0.0f) ? rsqrtf(fmaxf(d, 1e-12f)) : 0.0f;
    }
}

// One thread per (edge, group of 4 dims): float4 gather + 4 float atomics.
__global__ void conv_kernel(const float* __restrict__ x_in, float* __restrict__ x_out,
                            const int* __restrict__ row, const int* __restrict__ col,
                            const float* __restrict__ dinv) {
    long long idx = (long long)blockIdx.x * blockDim.x + threadIdx.x;
    if (idx >= (long long)N_EDGES * (DIM / 4)) return;
    int e = (int)(idx >> 4);
    int g = (int)(idx & 15);
    int r = row[e], c = col[e];
    float nrm = dinv[r] * dinv[c];
    const float4* src = (const float4*)(x_in + (long long)r * DIM);
    float4 xv = src[g];
    float* dst = x_out + (long long)c * DIM + g * 4;
    atomicAdd(dst + 0, nrm * xv.x);
    atomicAdd(dst + 1, nrm * xv.y);
    atomicAdd(dst + 2, nrm * xv.z);
    atomicAdd(dst + 3, nrm * xv.w);
}

// ------------------------------------------------------ fused WMMA node pass

#define WAVES_PER_BLOCK 4
#define TS 66   // LDS tile row stride in floats (even, reduces bank conflicts)

// D[16x16] tile of X @ W^T, K = 64, using V_WMMA_F32_16X16X4_F32 (16 steps).
// A 16x4 fp32 layout: lane l holds A[l%16][4*ks + 2*(l/16) + {0,1}].
// B 4x16 fp32 layout (mirror): lane l holds B[4*ks + 2*(l/16) + {0,1}][l%16];
// B[k][n] = W[n][k] for X@W^T.
__device__ inline v8f wmma_xwt_tile(const float* __restrict__ A, int lda,
                                    const float* __restrict__ W,
                                    int ncol_base, int lane) {
    v8f c = {0.f, 0.f, 0.f, 0.f, 0.f, 0.f, 0.f, 0.f};
    const int m    = lane & 15;
    const int koff = (lane >> 4) << 1;
    const float* wrow = W + (ncol_base + m) * DIM;   // output column n = ncol_base + m
    #pragma unroll
    for (int ks = 0; ks < 16; ++ks) {
        int k = ks * 4 + koff;
        v2f a, b;
        a.x = A[m * lda + k];
        a.y = A[m * lda + k + 1];
        b.x = wrow[k];
        b.y = wrow[k + 1];
        c = __builtin_amdgcn_wmma_f32_16x16x4_f32(false, a, false, b,
                                                  (short)0, c, false, false);
    }
    return c;
}

__global__ __launch_bounds__(32 * WAVES_PER_BLOCK)
void fused_node_kernel(const float* __restrict__ E,  const float* __restrict__ E2,
                       const float* __restrict__ x1, const float* __restrict__ x2,
                       const float* __restrict__ W0, const float* __restrict__ b0,
                       const float* __restrict__ W1, const float* __restrict__ b1,
                       const float* __restrict__ attnW, const float* __restrict__ attnb,
                       const float* __restrict__ qW,
                       float* __restrict__ Z) {
    __shared__ float zp_s[WAVES_PER_BLOCK][16 * TS];
    __shared__ float ba_s[WAVES_PER_BLOCK][16 * TS];
    __shared__ float bb_s[WAVES_PER_BLOCK][16 * TS];
    __shared__ float wsc_s[WAVES_PER_BLOCK][32];

    const int lane = threadIdx.x & 31;
    const int wave = threadIdx.x >> 5;
    const int NT   = NUM_NODES / 16;                 // 9375 (exact)
    int tile = blockIdx.x * WAVES_PER_BLOCK + wave;
    const bool active = (tile < NT);
    if (!active) tile = NT - 1;                      // clamp: keep EXEC all-1s for WMMA
    const long long base = (long long)tile * 16;

    float* zp   = zp_s[wave];
    float* bufA = ba_s[wave];
    float* bufB = bb_s[wave];
    float* wsc  = wsc_s[wave];

    const int m  = lane & 15;
    const int mh = (lane >> 4) * 8;                  // C/D row base for this half-wave

    // stage z_p = (E + x1 + x2)/3 and the E2 tile
    for (int i = lane; i < 16 * DIM; i += 32) {
        int r = i >> 6, d = i & 63;
        long long gi = (base + r) * DIM + d;
        zp[r * TS + d]   = (E[gi] + x1[gi] + x2[gi]) * (1.0f / 3.0f);
        bufA[r * TS + d] = E2[gi];
    }
    __syncthreads();

    // GEMM1: h = relu(E2 @ W0^T + b0) -> bufB
    for (int nt = 0; nt < 4; ++nt) {
        v8f c = wmma_xwt_tile(bufA, TS, W0, nt * 16, lane);
        int n = nt * 16 + m;
        float bias = b0[n];
        #pragma unroll
        for (int v = 0; v < 8; ++v) {
            float h = c[v] + bias;
            bufB[(mh + v) * TS + n] = (h > 0.0f) ? h : 0.0f;
        }
    }
    __syncthreads();

    // GEMM2: z_n = relu(h @ W1^T + b1) -> bufA
    for (int nt = 0; nt < 4; ++nt) {
        v8f c = wmma_xwt_tile(bufB, TS, W1, nt * 16, lane);
        int n = nt * 16 + m;
        float bias = b1[n];
        #pragma unroll
        for (int v = 0; v < 8; ++v) {
            float h = c[v] + bias;
            bufA[(mh + v) * TS + n] = (h > 0.0f) ? h : 0.0f;
        }
    }
    __syncthreads();

    float w_p, w_n;

    // attn on z_p: t = tanh(z_p @ attnW^T + attnb) -> bufB ; w_p = t . q
    for (int nt = 0; nt < 4; ++nt) {
        v8f c = wmma_xwt_tile(zp, TS, attnW, nt * 16, lane);
        int n = nt * 16 + m;
        float bias = attnb[n];
        #pragma unroll
        for (int v = 0; v < 8; ++v)
            bufB[(mh + v) * TS + n] = tanhf(c[v] + bias);
    }
    __syncthreads();
    {
        int d0 = (lane >> 4) * 32;
        float part = 0.0f;
        #pragma unroll
        for (int d = 0; d < 32; ++d) part += bufB[m * TS + d0 + d] * qW[d0 + d];
        w_p = part + __shfl_xor(part, 16, 32);
    }
    __syncthreads();

    // attn on z_n (bufA): -> bufB ; w_n = t . q
    for (int nt = 0; nt < 4; ++nt) {
        v8f c = wmma_xwt_tile(bufA, TS, attnW, nt * 16, lane);
        int n = nt * 16 + m;
        float bias = attnb[n];
        #pragma unroll
        for (int v = 0; v < 8; ++v)
            bufB[(mh + v) * TS + n] = tanhf(c[v] + bias);
    }
    __syncthreads();
    {
        int d0 = (lane >> 4) * 32;
        float part = 0.0f;
        #pragma unroll
        for (int d = 0; d < 32; ++d) part += bufB[m * TS + d0 + d] * qW[d0 + d];
        w_n = part + __shfl_xor(part, 16, 32);
    }

    if (lane < 16) { wsc[lane] = w_p; wsc[16 + lane] = w_n; }
    __syncthreads();

    // softmax fuse + store Z
    for (int i = lane; i < 16 * DIM; i += 32) {
        int r = i >> 6, d = i & 63;
        float wp = wsc[r], wn = wsc[16 + r];
        float mx = fmaxf(wp, wn);
        float ep = __expf(wp - mx), en = __expf(wn - mx);
        float ap = ep / (ep + en);
        float z  = ap * zp[r * TS + d] + (1.0f - ap) * bufA[r * TS + d];
        if (active) Z[(base + r) * DIM + d] = z;
    }
}

// ------------------------------------------------------------- SBPR loss

__device__ inline float wave_sum(float x) {
    #pragma unroll
    for (int s = 16; s >= 1; s >>= 1) x += __shfl_xor(x, s, 32);
    return x;
}

__global__ __launch_bounds__(32)
void loss_kernel(const float* __restrict__ Z, const float* __restrict__ w,
                 const int* __restrict__ u, const int* __restrict__ v,
                 const int* __restrict__ n, float* __restrict__ out) {
    int b = blockIdx.x;
    int lane = threadIdx.x;
    int d0 = lane, d1 = lane + 32;

    const float* ur = Z + (long long)u[b] * DIM;
    const float* vr = Z + (long long)v[b] * DIM;
    float u0 = ur[d0], u1 = ur[d1];
    float v0 = vr[d0], v1 = vr[d1];

    float pos = wave_sum(u0 * v0 + u1 * v1);
    float reg = u0 * u0 + u1 * u1 + v0 * v0 + v1 * v1;   // per-lane, reduced at end

    float wb = w[b];
    float sgn = (wb > 0.0f) ? 1.0f : ((wb < 0.0f) ? -1.0f : 0.0f);
    float cp = (-0.5f * sgn + 1.5f) * pos;

    float sbpr_sum = 0.0f;
    for (int k = 0; k < N_NEGS; ++k) {
        const float* nr = Z + (long long)n[b * N_NEGS + k] * DIM;
        float n0 = nr[d0], n1 = nr[d1];
        reg += n0 * n0 + n1 * n1;
        float neg = wave_sum(u0 * n0 + u1 * n1);
        float x = cp - neg;
        // stable log_sigmoid
        sbpr_sum += fminf(x, 0.0f) - log1pf(__expf(-fabsf(x)));
    }
    float reg_tot = wave_sum(reg);
    if (lane == 0) atomicAdd(out, -sbpr_sum + REG_C * reg_tot);
}

// ---------------------------------------------------------------- launcher

extern "C" void kernel_launch(void* const* d_in, const int* in_sizes, int n_in,
                              void* d_out, int out_size, void* d_ws, size_t ws_size,
                              hipStream_t stream) {
    const float* E     = (const float*)d_in[0];
    const float* E2    = (const float*)d_in[1];
    const float* W0    = (const float*)d_in[2];
    const float* b0    = (const float*)d_in[3];
    const float* W1    = (const float*)d_in[4];
    const float* b1    = (const float*)d_in[5];
    const float* attnW = (const float*)d_in[6];
    const float* attnb = (const float*)d_in[7];
    const float* qW    = (const float*)d_in[8];
    const float* wv    = (const float*)d_in[9];
    const int*   edge  = (const int*)d_in[10];
    const int*   u     = (const int*)d_in[11];
    const int*   v     = (const int*)d_in[12];
    const int*   nn    = (const int*)d_in[13];
    float* out = (float*)d_out;

    float* ws   = (float*)d_ws;
    float* deg  = ws;                                      // [N]
    float* dinv = ws + NUM_NODES;                          // [N]
    float* x1   = ws + 2LL * NUM_NODES;                    // [N*64]
    float* x2   = x1 + (long long)NUM_NODES * DIM;         // [N*64]
    float* Z    = x2 + (long long)NUM_NODES * DIM;         // [N*64]

    const int* row = edge;
    const int* col = edge + N_EDGES;

    // zero atomic accumulation targets (deg, x1, x2 contiguous pair) and output
    fill_zero_kernel<<<512, 256, 0, stream>>>(deg, NUM_NODES);
    fill_zero_kernel<<<4096, 256, 0, stream>>>(x1, 2LL * NUM_NODES * DIM);
    fill_zero_kernel<<<1, 32, 0, stream>>>(out, 1);

    deg_kernel<<<(N_EDGES + 255) / 256, 256, 0, stream>>>(col, deg);
    dinv_kernel<<<(NUM_NODES + 255) / 256, 256, 0, stream>>>(deg, dinv);

    long long cwork = (long long)N_EDGES * (DIM / 4);
    int cblocks = (int)((cwork + 255) / 256);
    conv_kernel<<<cblocks, 256, 0, stream>>>(E,  x1, row, col, dinv);
    conv_kernel<<<cblocks, 256, 0, stream>>>(x1, x2, row, col, dinv);

    int ntiles = NUM_NODES / 16;                                  // 9375
    int fblocks = (ntiles + WAVES_PER_BLOCK - 1) / WAVES_PER_BLOCK;
    fused_node_kernel<<<fblocks, 32 * WAVES_PER_BLOCK, 0, stream>>>(
        E, E2, x1, x2, W0, b0, W1, b1, attnW, attnb, qW, Z);

    loss_kernel<<<BATCH, 32, 0, stream>>>(Z, wv, u, v, nn, out);
}